// WilsonDslash_48739288875227
// MI455X (gfx1250) — compile-verified
//
#include <hip/hip_runtime.h>
#include <cstdint>

constexpr int LT = 32, LZ = 32, LY = 32, LX = 32;
constexpr int BY = 8;                  // y-rows per block
constexpr int ROWS = BY + 2;           // +halo
constexpr int SITE_F = 24;             // floats per psi site (4 spin x 3 color, complex)
constexpr int USITE_F = 72;            // floats per U site (4 mu x 9, complex)

struct cplx { float re, im; };

__device__ __forceinline__ void cmadd(cplx& a, const cplx& u, const cplx& p) {   // a += u*p
    a.re = __builtin_fmaf(u.re, p.re, __builtin_fmaf(-u.im, p.im, a.re));
    a.im = __builtin_fmaf(u.re, p.im, __builtin_fmaf( u.im, p.re, a.im));
}
__device__ __forceinline__ void cmaddc(cplx& a, const cplx& u, const cplx& p) {  // a += conj(u)*p
    a.re = __builtin_fmaf(u.re, p.re, __builtin_fmaf( u.im, p.im, a.re));
    a.im = __builtin_fmaf(u.re, p.im, __builtin_fmaf(-u.im, p.re, a.im));
}

__device__ __forceinline__ void unpack12(const float4 v[6], cplx s[12]) {
#pragma unroll
    for (int k = 0; k < 6; ++k) {
        s[2*k+0].re = v[k].x; s[2*k+0].im = v[k].y;
        s[2*k+1].re = v[k].z; s[2*k+1].im = v[k].w;
    }
}

__device__ __forceinline__ void load_psi_g(const float* __restrict__ psi, size_t site, cplx s[12]) {
    const float4* p = (const float4*)(psi + site * SITE_F);   // 96B-aligned
    float4 v[6];
#pragma unroll
    for (int k = 0; k < 6; ++k) v[k] = p[k];
    unpack12(v, s);
}

__device__ __forceinline__ void load_psi_lds(const float* tile, int cell, cplx s[12]) {
    const float4* p = (const float4*)(tile + cell * SITE_F);  // ds_load_b128
    float4 v[6];
#pragma unroll
    for (int k = 0; k < 6; ++k) v[k] = p[k];
    unpack12(v, s);
}

// U record for (site, MU): 18 floats at float-offset site*72 + MU*18.
// MU even -> 16B aligned (4x b128 + b64); MU odd -> 8B aligned (b64 + 4x b128).
template<int MU>
__device__ __forceinline__ void load_U9(const float* __restrict__ Ub, size_t site, cplx u[9]) {
    const float* base = Ub + site * USITE_F + MU * 18;
    if constexpr ((MU & 1) == 0) {
        const float4* q = (const float4*)base;
        float4 a = q[0], b = q[1], c = q[2], d = q[3];
        float2 e = *(const float2*)(base + 16);
        u[0]={a.x,a.y}; u[1]={a.z,a.w}; u[2]={b.x,b.y}; u[3]={b.z,b.w};
        u[4]={c.x,c.y}; u[5]={c.z,c.w}; u[6]={d.x,d.y}; u[7]={d.z,d.w}; u[8]={e.x,e.y};
    } else {
        float2 e = *(const float2*)base;
        const float4* q = (const float4*)(base + 2);
        float4 a = q[0], b = q[1], c = q[2], d = q[3];
        u[0]={e.x,e.y};
        u[1]={a.x,a.y}; u[2]={a.z,a.w}; u[3]={b.x,b.y}; u[4]={b.z,b.w};
        u[5]={c.x,c.y}; u[6]={c.z,c.w}; u[7]={d.x,d.y}; u[8]={d.z,d.w};
    }
}

// Forward hop: contrib = (I - g_mu) * (U * psi(x+mu)); half-spinor form.
template<int MU>
__device__ __forceinline__ void accum_fwd(cplx res[12], const cplx u[9], const cplx p[12]) {
    cplx h0[3], h1[3];
#pragma unroll
    for (int c = 0; c < 3; ++c) {
        const cplx p0 = p[c], p1 = p[3+c], p2 = p[6+c], p3 = p[9+c];
        if constexpr (MU == 0) {        // h0 = p0 - i p3 ; h1 = p1 - i p2
            h0[c] = {p0.re + p3.im, p0.im - p3.re};
            h1[c] = {p1.re + p2.im, p1.im - p2.re};
        } else if constexpr (MU == 1) { // h0 = p0 + p3 ; h1 = p1 - p2
            h0[c] = {p0.re + p3.re, p0.im + p3.im};
            h1[c] = {p1.re - p2.re, p1.im - p2.im};
        } else if constexpr (MU == 2) { // h0 = p0 - i p2 ; h1 = p1 + i p3
            h0[c] = {p0.re + p2.im, p0.im - p2.re};
            h1[c] = {p1.re - p3.im, p1.im + p3.re};
        } else {                        // h0 = p0 - p2 ; h1 = p1 - p3
            h0[c] = {p0.re - p2.re, p0.im - p2.im};
            h1[c] = {p1.re - p3.re, p1.im - p3.im};
        }
    }
#pragma unroll
    for (int i = 0; i < 3; ++i) {
        cplx uh0{0.f,0.f}, uh1{0.f,0.f};
#pragma unroll
        for (int j = 0; j < 3; ++j) { cmadd(uh0, u[3*i+j], h0[j]); cmadd(uh1, u[3*i+j], h1[j]); }
        res[i].re   += uh0.re; res[i].im   += uh0.im;
        res[3+i].re += uh1.re; res[3+i].im += uh1.im;
        if constexpr (MU == 0) {        // r2 += i*uh1 ; r3 += i*uh0
            res[6+i].re -= uh1.im; res[6+i].im += uh1.re;
            res[9+i].re -= uh0.im; res[9+i].im += uh0.re;
        } else if constexpr (MU == 1) { // r2 -= uh1 ; r3 += uh0
            res[6+i].re -= uh1.re; res[6+i].im -= uh1.im;
            res[9+i].re += uh0.re; res[9+i].im += uh0.im;
        } else if constexpr (MU == 2) { // r2 += i*uh0 ; r3 -= i*uh1
            res[6+i].re -= uh0.im; res[6+i].im += uh0.re;
            res[9+i].re += uh1.im; res[9+i].im -= uh1.re;
        } else {                        // r2 -= uh0 ; r3 -= uh1
            res[6+i].re -= uh0.re; res[6+i].im -= uh0.im;
            res[9+i].re -= uh1.re; res[9+i].im -= uh1.im;
        }
    }
}

// Backward hop: contrib = (I + g_mu) * (U(x-mu)^dag * psi(x-mu)).
template<int MU>
__device__ __forceinline__ void accum_bwd(cplx res[12], const cplx u[9], const cplx p[12]) {
    cplx h0[3], h1[3];
#pragma unroll
    for (int c = 0; c < 3; ++c) {
        const cplx p0 = p[c], p1 = p[3+c], p2 = p[6+c], p3 = p[9+c];
        if constexpr (MU == 0) {        // h0 = p0 + i p3 ; h1 = p1 + i p2
            h0[c] = {p0.re - p3.im, p0.im + p3.re};
            h1[c] = {p1.re - p2.im, p1.im + p2.re};
        } else if constexpr (MU == 1) { // h0 = p0 - p3 ; h1 = p1 + p2
            h0[c] = {p0.re - p3.re, p0.im - p3.im};
            h1[c] = {p1.re + p2.re, p1.im + p2.im};
        } else if constexpr (MU == 2) { // h0 = p0 + i p2 ; h1 = p1 - i p3
            h0[c] = {p0.re - p2.im, p0.im + p2.re};
            h1[c] = {p1.re + p3.im, p1.im - p3.re};
        } else {                        // h0 = p0 + p2 ; h1 = p1 + p3
            h0[c] = {p0.re + p2.re, p0.im + p2.im};
            h1[c] = {p1.re + p3.re, p1.im + p3.im};
        }
    }
#pragma unroll
    for (int i = 0; i < 3; ++i) {
        cplx uh0{0.f,0.f}, uh1{0.f,0.f};
#pragma unroll
        for (int j = 0; j < 3; ++j) { cmaddc(uh0, u[3*j+i], h0[j]); cmaddc(uh1, u[3*j+i], h1[j]); }
        res[i].re   += uh0.re; res[i].im   += uh0.im;
        res[3+i].re += uh1.re; res[3+i].im += uh1.im;
        if constexpr (MU == 0) {        // r2 -= i*uh1 ; r3 -= i*uh0
            res[6+i].re += uh1.im; res[6+i].im -= uh1.re;
            res[9+i].re += uh0.im; res[9+i].im -= uh0.re;
        } else if constexpr (MU == 1) { // r2 += uh1 ; r3 -= uh0
            res[6+i].re += uh1.re; res[6+i].im += uh1.im;
            res[9+i].re -= uh0.re; res[9+i].im -= uh0.im;
        } else if constexpr (MU == 2) { // r2 -= i*uh0 ; r3 += i*uh1
            res[6+i].re += uh0.im; res[6+i].im -= uh0.re;
            res[9+i].re -= uh1.im; res[9+i].im += uh1.re;
        } else {                        // r2 += uh0 ; r3 += uh1
            res[6+i].re += uh0.re; res[6+i].im += uh0.im;
            res[9+i].re += uh1.re; res[9+i].im += uh1.im;
        }
    }
}

__global__ __launch_bounds__(256) void wilson_dslash_kernel(const float* __restrict__ psi,
                                                            const float* __restrict__ U,
                                                            float* __restrict__ out) {
    __shared__ float tile[ROWS * LX * SITE_F];   // 10*32*96B = 30 KB psi tile (y-halo)

    const int x  = threadIdx.x;
    const int ty = threadIdx.y;
    const int y0 = blockIdx.x * BY;
    const int z  = blockIdx.y;
    const int t  = blockIdx.z;
    const int y  = y0 + ty;
    const int tid = ty * LX + x;

    const size_t plane = ((size_t)t * LZ + z) * LY;   // (t,z) row base in y-units

    // ---- Phase 0: async-stage psi rows [y0-1 .. y0+BY] into LDS (CDNA5 async DMA) ----
    // One whole 96-B site per (thread, iteration): 1 global base + 1 LDS base per cell,
    // the six b128 chunks are addressed purely via the instruction offset immediate
    // (INST_OFFSET is added to BOTH the LDS and global addresses for async loads).
    for (int cell = tid; cell < ROWS * LX; cell += 256) {      // 320 cells, 1-2 per thread
        const int r  = cell >> 5;                              // tile row
        const int xx = cell & (LX - 1);
        const int gy = (y0 + r + LY - 1) & (LY - 1);
        uint64_t gaddr = (uint64_t)(uintptr_t)(psi + ((plane + gy) * LX + xx) * SITE_F);
        uint32_t laddr = (uint32_t)(uintptr_t)(&tile[cell * SITE_F]);
        asm volatile("global_load_async_to_lds_b128 %0, %1, off\n\t"
                     "global_load_async_to_lds_b128 %0, %1, off offset:16\n\t"
                     "global_load_async_to_lds_b128 %0, %1, off offset:32\n\t"
                     "global_load_async_to_lds_b128 %0, %1, off offset:48\n\t"
                     "global_load_async_to_lds_b128 %0, %1, off offset:64\n\t"
                     "global_load_async_to_lds_b128 %0, %1, off offset:80"
                     :: "v"(laddr), "v"(gaddr) : "memory");
    }

    cplx res[12];
#pragma unroll
    for (int k = 0; k < 12; ++k) res[k] = {0.f, 0.f};

    const size_t site = (plane + y) * LX + x;
    cplx pn[12], u[9];

    // ---- Phase 1: T and Z directions from global/L2 (overlaps the async LDS fill) ----
    {   // mu = 0 : T
        size_t sf = (((size_t)((t + 1) & (LT - 1)) * LZ + z) * LY + y) * LX + x;
        load_psi_g(psi, sf, pn);
        load_U9<0>(U, site, u);
        accum_fwd<0>(res, u, pn);
        size_t sb = (((size_t)((t + LT - 1) & (LT - 1)) * LZ + z) * LY + y) * LX + x;
        load_psi_g(psi, sb, pn);
        load_U9<0>(U, sb, u);
        accum_bwd<0>(res, u, pn);
    }
    {   // mu = 1 : Z
        size_t sf = (((size_t)t * LZ + ((z + 1) & (LZ - 1))) * LY + y) * LX + x;
        load_psi_g(psi, sf, pn);
        load_U9<1>(U, site, u);
        accum_fwd<1>(res, u, pn);
        size_t sb = (((size_t)t * LZ + ((z + LZ - 1) & (LZ - 1))) * LY + y) * LX + x;
        load_psi_g(psi, sb, pn);
        load_U9<1>(U, sb, u);
        accum_bwd<1>(res, u, pn);
    }

    // ---- Drain async DMA, make tile visible to all 8 waves ----
    asm volatile("s_wait_asynccnt 0x0" ::: "memory");
    __syncthreads();

    // ---- Phase 2: Y and X directions, psi from LDS ----
    {   // mu = 2 : Y   (tile row r <-> global y0-1+r; own row = ty+1)
        load_psi_lds(tile, (ty + 2) * LX + x, pn);
        load_U9<2>(U, site, u);
        accum_fwd<2>(res, u, pn);
        size_t sb = (plane + ((y + LY - 1) & (LY - 1))) * LX + x;
        load_psi_lds(tile, (ty + 0) * LX + x, pn);
        load_U9<2>(U, sb, u);
        accum_bwd<2>(res, u, pn);
    }
    {   // mu = 3 : X   (x wraps inside the row)
        load_psi_lds(tile, (ty + 1) * LX + ((x + 1) & (LX - 1)), pn);
        load_U9<3>(U, site, u);
        accum_fwd<3>(res, u, pn);
        size_t sb = (plane + y) * LX + ((x + LX - 1) & (LX - 1));
        load_psi_lds(tile, (ty + 1) * LX + ((x + LX - 1) & (LX - 1)), pn);
        load_U9<3>(U, sb, u);
        accum_bwd<3>(res, u, pn);
    }

    // ---- Store: out = -0.5 * sum  (6 x b128) ----
    float4* o = (float4*)(out + site * SITE_F);
#pragma unroll
    for (int k = 0; k < 6; ++k) {
        o[k] = make_float4(-0.5f * res[2*k].re,   -0.5f * res[2*k].im,
                           -0.5f * res[2*k+1].re, -0.5f * res[2*k+1].im);
    }
}

extern "C" void kernel_launch(void* const* d_in, const int* in_sizes, int n_in,
                              void* d_out, int out_size, void* d_ws, size_t ws_size,
                              hipStream_t stream) {
    (void)in_sizes; (void)n_in; (void)out_size; (void)d_ws; (void)ws_size;
    const float* psi = (const float*)d_in[0];
    const float* U   = (const float*)d_in[1];
    float* out = (float*)d_out;

    dim3 grid(LY / BY, LZ, LT);   // (4, 32, 32)
    dim3 block(LX, BY);           // 32 x 8 = 256 threads = 8 wave32
    hipLaunchKernelGGL(wilson_dslash_kernel, grid, block, 0, stream, psi, U, out);
}